// NodeClassifier_28398323761928
// MI455X (gfx1250) — compile-verified
//
#include <hip/hip_runtime.h>
#include <math.h>

// ---------------------------------------------------------------------------
// Hetero GATv2 (2 layers, 4 edge types, H=2 heads x C=64) + MLP head, MI455X.
//
// Dense transforms use V_WMMA_F32_16X16X4_F32 (full fp32 precision, chained
// K=4 steps over K=128). Edge phase is L2-resident (XL/XR/MSG = 25.6MB each
// vs 192MB L2) so atomic scatter-add is the right data-movement path.
// ---------------------------------------------------------------------------

#define NN       50000
#define EE_EDGES 400000
#define TT       4
#define F_INN    128
#define HH       2
#define CC       64
#define HCC      128
#define MLPH     8
#define NCLS     10

typedef float v2f __attribute__((ext_vector_type(2)));
typedef float v8f __attribute__((ext_vector_type(8)));

// ---------------------------------------------------------------------------
// GEMM: O[rowTile*16.., colTile*16..] = H(50000x128) @ W(128x128)
// One wave per 16x16 tile; 8 waves/block = 8 column tiles; grid.y picks Wl/Wr.
// ---------------------------------------------------------------------------
__global__ __launch_bounds__(256) void gemm_xlxr_kernel(
    const float* __restrict__ Hin,
    const float* __restrict__ Wl, const float* __restrict__ Wr,
    float* __restrict__ XL, float* __restrict__ XR)
{
    const int lane    = threadIdx.x & 31;
    const int wave    = threadIdx.x >> 5;       // column tile 0..7
    const int rowTile = blockIdx.x;             // 0..3124
    const float* __restrict__ W = (blockIdx.y == 0) ? Wl : Wr;
    float* __restrict__ O       = (blockIdx.y == 0) ? XL : XR;

    const int arow  = rowTile * 16 + (lane & 15);   // A row this lane feeds
    const int bcol  = wave * 16 + (lane & 15);      // B col this lane feeds
    const int khalf = (lane >> 4) * 2;              // 0 (lanes 0-15) or 2

    v8f acc = {0.f, 0.f, 0.f, 0.f, 0.f, 0.f, 0.f, 0.f};

#pragma unroll
    for (int kb = 0; kb < 32; ++kb) {
        const int k = kb * 4 + khalf;               // even -> 8B aligned
        // A: 16x4 fp32 tile, lane holds (M=arow, K=k,k+1)
        v2f a = *(const v2f*)(Hin + arow * F_INN + k);
        // B: 4x16 fp32 tile, lane holds (K=k,k+1 ; N=bcol)
        v2f b;
        b.x = W[k * HCC + bcol];
        b.y = W[(k + 1) * HCC + bcol];
        acc = __builtin_amdgcn_wmma_f32_16x16x4_f32(
            /*neg_a=*/false, a, /*neg_b=*/false, b,
            /*c_mod=*/(short)0, acc, /*reuse_a=*/false, /*reuse_b=*/false);
    }

    // C/D layout: lanes 0-15 hold M=0..7 (vgpr i), lanes 16-31 hold M=8..15
    const int mbase = rowTile * 16 + (lane >> 4) * 8;
    const int ncol  = wave * 16 + (lane & 15);
#pragma unroll
    for (int i = 0; i < 8; ++i)
        O[(mbase + i) * HCC + ncol] = acc[i];
}

// ---------------------------------------------------------------------------
// Helpers / small kernels
// ---------------------------------------------------------------------------
__device__ __forceinline__ float lrelu(float v) { return v > 0.f ? v : 0.2f * v; }

__device__ __forceinline__ void atomicMaxF32(float* addr, float val)
{
    // sign-split trick; addr initialized to -inf
    if (val >= 0.f) atomicMax((int*)addr, __float_as_int(val));
    else            atomicMin((unsigned int*)addr, __float_as_uint(val));
}

__global__ void init_seg_kernel(float* __restrict__ smax, float* __restrict__ denom, int n)
{
    int i = blockIdx.x * blockDim.x + threadIdx.x;
    if (i < n) { smax[i] = -INFINITY; denom[i] = 0.f; }
}

__global__ void zero_kernel(float* __restrict__ p, int n)
{
    int i = blockIdx.x * blockDim.x + threadIdx.x;
    if (i < n) p[i] = 0.f;
}

__global__ void relu_kernel(float* __restrict__ p, int n)
{
    int i = blockIdx.x * blockDim.x + threadIdx.x;
    if (i < n) p[i] = fmaxf(p[i], 0.f);
}

// ---------------------------------------------------------------------------
// Edge pass 1: per-edge attention score + segment-max over destinations.
// One wave per edge; lanes 0-15 = head 0 (chs 0-63), lanes 16-31 = head 1.
// ---------------------------------------------------------------------------
__global__ __launch_bounds__(256) void edge_score_kernel(
    const int* __restrict__ src, const int* __restrict__ dst,
    const float* __restrict__ XL, const float* __restrict__ XR,
    const float* __restrict__ att,   // [H*C] for this type
    float* __restrict__ score,       // [(E+N)*H]
    float* __restrict__ smax)        // [N*H]
{
    const int e = blockIdx.x * (blockDim.x >> 5) + (threadIdx.x >> 5);
    const int EE = EE_EDGES + NN;
    if (e >= EE) return;
    const int s = (e < EE_EDGES) ? src[e] : (e - EE_EDGES);
    const int d = (e < EE_EDGES) ? dst[e] : (e - EE_EDGES);

    const int lane = threadIdx.x & 31;
    const int ch   = lane * 4;
    float4 xl = *(const float4*)(XL + s * HCC + ch);
    float4 xr = *(const float4*)(XR + d * HCC + ch);
    float4 av = *(const float4*)(att + ch);

    float sum = lrelu(xl.x + xr.x) * av.x
              + lrelu(xl.y + xr.y) * av.y
              + lrelu(xl.z + xr.z) * av.z
              + lrelu(xl.w + xr.w) * av.w;

    // reduce within each 16-lane half (one head per half)
#pragma unroll
    for (int off = 1; off < 16; off <<= 1)
        sum += __shfl_xor(sum, off, 32);

    if ((lane & 15) == 0) {
        const int h = lane >> 4;
        score[e * HH + h] = sum;
        atomicMaxF32(&smax[d * HH + h], sum);
    }
}

// ---------------------------------------------------------------------------
// Edge pass 2: p = exp(score - smax[d]); denom[d] += p; msg[d] += p * xl[s].
// ---------------------------------------------------------------------------
__global__ __launch_bounds__(256) void edge_accum_kernel(
    const int* __restrict__ src, const int* __restrict__ dst,
    const float* __restrict__ XL,
    const float* __restrict__ score, const float* __restrict__ smax,
    float* __restrict__ denom,       // [N*H]
    float* __restrict__ msg)         // [N*HC]
{
    const int e = blockIdx.x * (blockDim.x >> 5) + (threadIdx.x >> 5);
    const int EE = EE_EDGES + NN;
    if (e >= EE) return;
    const int s = (e < EE_EDGES) ? src[e] : (e - EE_EDGES);
    const int d = (e < EE_EDGES) ? dst[e] : (e - EE_EDGES);

    const int lane = threadIdx.x & 31;
    const int h    = lane >> 4;
    const float p  = __expf(score[e * HH + h] - smax[d * HH + h]);

    if ((lane & 15) == 0) atomicAdd(&denom[d * HH + h], p);

    const int ch = lane * 4;
    float4 xl = *(const float4*)(XL + s * HCC + ch);
    float* m = msg + d * HCC + ch;
    atomicAdd(m + 0, p * xl.x);
    atomicAdd(m + 1, p * xl.y);
    atomicAdd(m + 2, p * xl.z);
    atomicAdd(m + 3, p * xl.w);
}

// ---------------------------------------------------------------------------
// Finalize one edge type: hout += msg / denom + bias
// ---------------------------------------------------------------------------
__global__ void finalize_kernel(
    const float* __restrict__ msg, const float* __restrict__ denom,
    const float* __restrict__ bias,  // [HC] for this type
    float* __restrict__ hout, int n) // n = N*HC
{
    int i = blockIdx.x * blockDim.x + threadIdx.x;
    if (i >= n) return;
    const int node = i >> 7;          // /128
    const int c    = i & 127;
    const int h    = c >> 6;          // /64
    hout[i] += msg[i] / denom[node * HH + h] + bias[c];
}

// ---------------------------------------------------------------------------
// MLP head: h(N x 128) -> relu(h@W1 + b1)(N x 8) -> @W2 + b2 (N x 10)
// ---------------------------------------------------------------------------
__global__ void mlp_head_kernel(
    const float* __restrict__ Hin,
    const float* __restrict__ W1, const float* __restrict__ B1,
    const float* __restrict__ W2, const float* __restrict__ B2,
    float* __restrict__ out)
{
    int node = blockIdx.x * blockDim.x + threadIdx.x;
    if (node >= NN) return;
    float hid[MLPH];
#pragma unroll
    for (int j = 0; j < MLPH; ++j) hid[j] = B1[j];
    const float* hrow = Hin + node * HCC;
    for (int k = 0; k < HCC; ++k) {
        float v = hrow[k];
#pragma unroll
        for (int j = 0; j < MLPH; ++j) hid[j] += v * W1[k * MLPH + j];
    }
#pragma unroll
    for (int j = 0; j < MLPH; ++j) hid[j] = fmaxf(hid[j], 0.f);
#pragma unroll
    for (int c = 0; c < NCLS; ++c) {
        float o = B2[c];
#pragma unroll
        for (int j = 0; j < MLPH; ++j) o += hid[j] * W2[j * NCLS + c];
        out[node * NCLS + c] = o;
    }
}

// ---------------------------------------------------------------------------
// Host launcher
// ---------------------------------------------------------------------------
extern "C" void kernel_launch(void* const* d_in, const int* in_sizes, int n_in,
                              void* d_out, int out_size, void* d_ws, size_t ws_size,
                              hipStream_t stream)
{
    const float* x     = (const float*)d_in[0];
    const int*   ei    = (const int*)  d_in[1];   // [T,2,E]
    const float* Wl1   = (const float*)d_in[2];
    const float* Wr1   = (const float*)d_in[3];
    const float* att1  = (const float*)d_in[4];
    const float* b1    = (const float*)d_in[5];
    const float* Wl2   = (const float*)d_in[6];
    const float* Wr2   = (const float*)d_in[7];
    const float* att2  = (const float*)d_in[8];
    const float* b2    = (const float*)d_in[9];
    const float* Wlin1 = (const float*)d_in[10];
    const float* blin1 = (const float*)d_in[11];
    const float* Wlin2 = (const float*)d_in[12];
    const float* blin2 = (const float*)d_in[13];

    // Workspace layout (floats): ~107 MB total
    float* ws    = (float*)d_ws;
    float* H1    = ws;                    // N*HC  layer-1 output
    float* H2    = H1 + (size_t)NN * HCC; // N*HC  layer-2 output
    float* XL    = H2 + (size_t)NN * HCC; // N*HC
    float* XR    = XL + (size_t)NN * HCC; // N*HC  (reused as MSG after pass 1)
    float* SMAX  = XR + (size_t)NN * HCC; // N*H
    float* DEN   = SMAX + (size_t)NN * HH;// N*H
    float* SCORE = DEN + (size_t)NN * HH; // (E+N)*H
    float* MSG   = XR;                    // alias: XR dead after edge pass 1

    const int nHC    = NN * HCC;
    const int EE     = EE_EDGES + NN;
    const int eBlks  = (EE + 7) / 8;             // 8 waves (edges) per 256-thr block
    const int hcBlks = (nHC + 255) / 256;
    const int segBlk = (NN * HH + 255) / 256;

    auto run_layer = [&](const float* hin, float* hout,
                         const float* Wl, const float* Wr,
                         const float* att, const float* bias) {
        zero_kernel<<<hcBlks, 256, 0, stream>>>(hout, nHC);
        for (int t = 0; t < TT; ++t) {
            const int* src_t = ei + (size_t)t * 2 * EE_EDGES;
            const int* dst_t = src_t + EE_EDGES;
            gemm_xlxr_kernel<<<dim3(NN / 16, 2), 256, 0, stream>>>(
                hin, Wl + (size_t)t * F_INN * HCC, Wr + (size_t)t * F_INN * HCC, XL, XR);
            init_seg_kernel<<<segBlk, 256, 0, stream>>>(SMAX, DEN, NN * HH);
            edge_score_kernel<<<eBlks, 256, 0, stream>>>(
                src_t, dst_t, XL, XR, att + t * HH * CC, SCORE, SMAX);
            zero_kernel<<<hcBlks, 256, 0, stream>>>(MSG, nHC);   // MSG aliases XR
            edge_accum_kernel<<<eBlks, 256, 0, stream>>>(
                src_t, dst_t, XL, SCORE, SMAX, DEN, MSG);
            finalize_kernel<<<hcBlks, 256, 0, stream>>>(
                MSG, DEN, bias + t * HCC, hout, nHC);
        }
        relu_kernel<<<hcBlks, 256, 0, stream>>>(hout, nHC);
    };

    run_layer(x,  H1, Wl1, Wr1, att1, b1);
    run_layer(H1, H2, Wl2, Wr2, att2, b2);

    mlp_head_kernel<<<(NN + 127) / 128, 128, 0, stream>>>(
        H2, Wlin1, blin1, Wlin2, blin2, (float*)d_out);
}